// GaussianPrimitives_59854664237411
// MI455X (gfx1250) — compile-verified
//
#include <hip/hip_runtime.h>
#include <cstdint>
#include <cstddef>

#define TPB 256

// Sigma = Q * diag(exp(2*log_s)) * Q^T  per point, Q from normalized quaternion.
// Streaming, bandwidth-bound: direct vec loads in, LDS-staged coalesced
// async (LDS -> global) b128 stores out.
__global__ __launch_bounds__(TPB, 2) void gaussian_sigma_kernel(
    const float* __restrict__ q,
    const float* __restrict__ log_s,
    float* __restrict__ out,
    int n)
{
    __shared__ float smem[TPB * 9];   // 9216 B output staging tile

    const int t  = threadIdx.x;
    const int p0 = blockIdx.x * TPB;  // first point handled by this block
    const int i  = p0 + t;

    float S00 = 0.f, S01 = 0.f, S02 = 0.f, S11 = 0.f, S12 = 0.f, S22 = 0.f;

    if (i < n) {
        const float4 qv = reinterpret_cast<const float4*>(q)[i];  // global_load_b128
        float w = qv.x, x = qv.y, y = qv.z, z = qv.w;
        const float l0 = log_s[3 * i + 0];
        const float l1 = log_s[3 * i + 1];
        const float l2 = log_s[3 * i + 2];

        const float inv = rsqrtf(w * w + x * x + y * y + z * z);
        w *= inv; x *= inv; y *= inv; z *= inv;

        const float xx = x * x, yy = y * y, zz = z * z;
        const float xy = x * y, xz = x * z, yz = y * z;
        const float wx = w * x, wy = w * y, wz = w * z;

        const float r00 = 1.f - 2.f * (yy + zz);
        const float r01 = 2.f * (xy - wz);
        const float r02 = 2.f * (xz + wy);
        const float r10 = 2.f * (xy + wz);
        const float r11 = 1.f - 2.f * (xx + zz);
        const float r12 = 2.f * (yz - wx);
        const float r20 = 2.f * (xz - wy);
        const float r21 = 2.f * (yz + wx);
        const float r22 = 1.f - 2.f * (xx + yy);

        const float s0 = __expf(2.f * l0);   // v_exp_f32
        const float s1 = __expf(2.f * l1);
        const float s2 = __expf(2.f * l2);

        // M = R * diag(s); Sigma = M * R^T (symmetric -> 6 unique terms)
        const float a0 = s0 * r00, a1 = s1 * r01, a2 = s2 * r02;
        const float b0 = s0 * r10, b1 = s1 * r11, b2 = s2 * r12;
        const float c0 = s0 * r20, c1 = s1 * r21, c2 = s2 * r22;

        S00 = a0 * r00 + a1 * r01 + a2 * r02;
        S01 = a0 * r10 + a1 * r11 + a2 * r12;
        S02 = a0 * r20 + a1 * r21 + a2 * r22;
        S11 = b0 * r10 + b1 * r11 + b2 * r12;
        S12 = b0 * r20 + b1 * r21 + b2 * r22;
        S22 = c0 * r20 + c1 * r21 + c2 * r22;
    }

    const bool full_block = (p0 + TPB) <= n;
    if (full_block) {
        // Stage: thread t owns dwords [t*9, t*9+9). Stride 9, gcd(9,64)=1 ->
        // bank-conflict-free across the wave.
        float* row = &smem[t * 9];
        row[0] = S00; row[1] = S01; row[2] = S02;
        row[3] = S01; row[4] = S11; row[5] = S12;
        row[6] = S02; row[7] = S12; row[8] = S22;
        __syncthreads();

        // Drain the 9216 B tile with 576 async b128 LDS->global chunks.
        // Block base byte offset = p0 * 36; 256*36 = 9216 is 16B-aligned.
        const uint64_t gbase = (uint64_t)(uintptr_t)out + (uint64_t)p0 * 36u;
        const uint32_t lbase = (uint32_t)(uintptr_t)(&smem[0]);

        {
            const uint64_t ga = gbase + (uint32_t)t * 16u;
            const uint32_t la = lbase + (uint32_t)t * 16u;
            asm volatile("global_store_async_from_lds_b128 %0, %1, off"
                         :: "v"(ga), "v"(la) : "memory");
        }
        {
            const uint64_t ga = gbase + (uint32_t)(t + TPB) * 16u;
            const uint32_t la = lbase + (uint32_t)(t + TPB) * 16u;
            asm volatile("global_store_async_from_lds_b128 %0, %1, off"
                         :: "v"(ga), "v"(la) : "memory");
        }
        if (t < (TPB * 9 / 4 - 2 * TPB)) {   // remaining 64 chunks
            const uint64_t ga = gbase + (uint32_t)(t + 2 * TPB) * 16u;
            const uint32_t la = lbase + (uint32_t)(t + 2 * TPB) * 16u;
            asm volatile("global_store_async_from_lds_b128 %0, %1, off"
                         :: "v"(ga), "v"(la) : "memory");
        }
        // Drain ASYNCcnt (s_endpgm also implies a full wait; this keeps the
        // LDS-lifetime requirement explicit).
        asm volatile("s_wait_asynccnt 0x0" ::: "memory");
    } else if (i < n) {
        // Tail block (not hit for N = 4M): plain guarded stores.
        float* o = out + (size_t)i * 9;
        o[0] = S00; o[1] = S01; o[2] = S02;
        o[3] = S01; o[4] = S11; o[5] = S12;
        o[6] = S02; o[7] = S12; o[8] = S22;
    }
}

extern "C" void kernel_launch(void* const* d_in, const int* in_sizes, int n_in,
                              void* d_out, int out_size, void* d_ws, size_t ws_size,
                              hipStream_t stream)
{
    const float* q     = (const float*)d_in[0];   // N x 4 fp32
    const float* log_s = (const float*)d_in[1];   // N x 3 fp32
    float*       out   = (float*)d_out;           // N x 3 x 3 fp32

    const int n      = in_sizes[0] / 4;           // number of points
    const int blocks = (n + TPB - 1) / TPB;       // 15625 for N = 4e6

    gaussian_sigma_kernel<<<blocks, TPB, 0, stream>>>(q, log_s, out, n);
}